// MPFIL_MutPred_34540126994711
// MI455X (gfx1250) — compile-verified
//
#include <hip/hip_runtime.h>
#include <hip/hip_bf16.h>
#include <math.h>
#include <cstdint>

// ---------------------------------------------------------------------------
// CDNA5 (gfx1250) full-network forward pass.
// All dense algebra goes through a tiled f32 WMMA GEMM using
// __builtin_amdgcn_wmma_f32_16x16x4_f32 (exact f32 path on the matrix pipes),
// with double-buffered LDS staging via the gfx1250 async copy path
// (global_load_async_to_lds_b128 + s_wait_asynccnt, via inline asm).
// ---------------------------------------------------------------------------

typedef float v2f __attribute__((ext_vector_type(2)));
typedef float v8f __attribute__((ext_vector_type(8)));

constexpr int EMB   = 1024;
constexpr int BSZ   = 16;
constexpr int SEQL  = 500;
constexpr int LSEG  = 499;
constexpr int NNODE = BSZ * LSEG;   // 7984
constexpr int NED   = 256000;
constexpr int NB3   = 48;           // BS*3
constexpr int NTOK  = NB3 * SEQL;   // 24000
constexpr int C1 = 1024, C2 = 2048;

constexpr int TBM = 64, TBN = 128;  // block tile
constexpr int LPAD = 20;            // LDS row stride (floats): 80B -> 16B-aligned rows

enum { A_NONE = 0, A_RELU = 1, A_SILU = 2, A_TANH = 3, A_SILU2 = 4 };

#if defined(__gfx1250__)
#define HAS_ASYNC_LDS 1
#else
#define HAS_ASYNC_LDS 0
#endif

#if HAS_ASYNC_LDS
__device__ __forceinline__ void async_cp16(void* lds_dst, const void* gsrc) {
    // VDST = LDS byte address (low 32 bits of flat LDS pointer);
    // VADDR = 64-bit global address; "off" = null SADDR (GV mode).
    uint32_t l = (uint32_t)(uintptr_t)lds_dst;
    uint64_t g = (uint64_t)(uintptr_t)gsrc;
    asm volatile("global_load_async_to_lds_b128 %0, %1, off"
                 :: "v"(l), "v"(g) : "memory");
}
__device__ __forceinline__ void async_wait0() {
    asm volatile("s_wait_asynccnt 0x0" ::: "memory");
}
#else
__device__ __forceinline__ void async_wait0() {}
#endif

__device__ __forceinline__ float act_apply(float v, int act) {
    if (act == A_RELU)  return fmaxf(v, 0.f);
    if (act == A_SILU)  return v / (1.f + expf(-v));
    if (act == A_TANH)  return tanhf(v);
    if (act == A_SILU2) { v = v / (1.f + expf(-v)); return v / (1.f + expf(-v)); }
    return v;
}

// ---------------------------------------------------------------------------
// WMMA f32 GEMM:  C = act( (A @ op(B) + bias) * alpha [*scale2 + bias2] [+res] )
//   A: [M,K] row-major, lda.   TB=1: B is W[N,K] (A@W^T).  TB=0: B is [K,N].
//   Batched via blockIdx.z -> (zo = z/nInner, zi = z%nInner) outer/inner strides.
// ---------------------------------------------------------------------------
template<int TB>
__global__ __launch_bounds__(256) void k_gemm(
    const float* __restrict__ A, long long lda, long long sAo, long long sAi,
    const float* __restrict__ B, long long ldb, long long sBo, long long sBi,
    float* __restrict__ C, long long ldc, long long sCo, long long sCi,
    const float* __restrict__ bias,
    const float* __restrict__ scale2, const float* __restrict__ bias2,
    const float* __restrict__ res, long long ldr,
    int M, int N, int K, int nInner, float alpha, int act)
{
    __shared__ float sA[2][TBM][LPAD];
    __shared__ float sB[2][TBN][LPAD];

    const int z  = blockIdx.z;
    const int zo = z / nInner, zi = z % nInner;
    A += (long long)zo * sAo + (long long)zi * sAi;
    B += (long long)zo * sBo + (long long)zi * sBi;
    C += (long long)zo * sCo + (long long)zi * sCi;

    const int m0 = blockIdx.y * TBM;
    const int n0 = blockIdx.x * TBN;
    const int tid  = threadIdx.x;
    const int lane = tid & 31;
    const int wave = tid >> 5;          // 8 waves: 2 (M) x 4 (N)
    const int wM = wave >> 2, wN = wave & 3;
    const int half = lane >> 4, l16 = lane & 15;

    v8f acc[2][2];
    for (int i = 0; i < 2; i++)
        for (int j = 0; j < 2; j++)
            for (int e = 0; e < 8; e++) acc[i][j][e] = 0.f;

    // ---- tile loaders (16-wide K slab) -----------------------------------
    auto loadA = [&](int k0, float (*dA)[LPAD]) {
        int t4 = tid * 4;
        int r  = t4 >> 4, c = t4 & 15;
        int gr = m0 + r;
        if (gr < M && (k0 + 16) <= K) {
            const float* src = &A[(long long)gr * lda + k0 + c];
#if HAS_ASYNC_LDS
            async_cp16(&dA[r][c], src);
#else
            const float4 v = *reinterpret_cast<const float4*>(src);
            dA[r][c] = v.x; dA[r][c+1] = v.y; dA[r][c+2] = v.z; dA[r][c+3] = v.w;
            if (k0 + 16 < K) __builtin_prefetch(&A[(long long)gr * lda + k0 + 16], 0, 1);
#endif
        } else {
            for (int j = 0; j < 4; j++) {
                int kk = k0 + c + j;
                dA[r][c + j] = (gr < M && kk < K) ? A[(long long)gr * lda + kk] : 0.f;
            }
        }
    };
    auto loadB = [&](int k0, float (*dB)[LPAD]) {
        if (TB) {
            for (int i = 0; i < 2; i++) {
                int idx = (tid + i * 256) * 4;
                int r = idx >> 4, c = idx & 15;
                int gn = n0 + r;
                if (gn < N && (k0 + 16) <= K) {
                    const float* src = &B[(long long)gn * ldb + k0 + c];
#if HAS_ASYNC_LDS
                    async_cp16(&dB[r][c], src);
#else
                    const float4 v = *reinterpret_cast<const float4*>(src);
                    dB[r][c] = v.x; dB[r][c+1] = v.y; dB[r][c+2] = v.z; dB[r][c+3] = v.w;
#endif
                } else {
                    for (int j = 0; j < 4; j++) {
                        int kk = k0 + c + j;
                        dB[r][c + j] = (gn < N && kk < K) ? B[(long long)gn * ldb + kk] : 0.f;
                    }
                }
            }
        } else {
            for (int i = 0; i < 2; i++) {
                int idx = tid + i * 256;          // 0..511
                int kk  = idx >> 5;               // 0..15
                int c4  = (idx & 31) * 4;         // 0..124
                int gk  = k0 + kk;
                for (int j = 0; j < 4; j++) {
                    int gn = n0 + c4 + j;
                    dB[c4 + j][kk] = (gk < K && gn < N) ? B[(long long)gk * ldb + gn] : 0.f;
                }
            }
        }
    };
    auto compute = [&](float (*cA)[LPAD], float (*cB)[LPAD]) {
        #pragma unroll
        for (int kk = 0; kk < 16; kk += 4) {
            v2f aF[2], bF[2];
            #pragma unroll
            for (int sm = 0; sm < 2; sm++) {
                int rr = wM * 32 + sm * 16 + l16;   // frag pair is contiguous: one ds_load_b64
                aF[sm] = *reinterpret_cast<const v2f*>(&cA[rr][kk + 2 * half]);
            }
            #pragma unroll
            for (int sn = 0; sn < 2; sn++) {
                int cc = wN * 32 + sn * 16 + l16;
                bF[sn] = *reinterpret_cast<const v2f*>(&cB[cc][kk + 2 * half]);
            }
            #pragma unroll
            for (int sm = 0; sm < 2; sm++)
                #pragma unroll
                for (int sn = 0; sn < 2; sn++)
                    acc[sm][sn] = __builtin_amdgcn_wmma_f32_16x16x4_f32(
                        false, aF[sm], false, bF[sn], (short)0, acc[sm][sn], false, false);
        }
    };

    // ---- double-buffered main loop ---------------------------------------
    const int nk = (K + 15) >> 4;
    loadA(0, sA[0]); loadB(0, sB[0]);
    async_wait0();
    __syncthreads();
    int cur = 0;
    for (int i = 0; i < nk; i++) {
        if (i + 1 < nk) { loadA((i + 1) << 4, sA[cur ^ 1]); loadB((i + 1) << 4, sB[cur ^ 1]); }
        compute(sA[cur], sB[cur]);
        async_wait0();
        __syncthreads();
        cur ^= 1;
    }

    // ---- epilogue
    for (int sm = 0; sm < 2; sm++)
        for (int sn = 0; sn < 2; sn++) {
            int cc = n0 + wN * 32 + sn * 16 + l16;
            if (cc >= N) continue;
            float bv = bias ? bias[cc] : 0.f;
            float s2 = scale2 ? scale2[cc] : 1.f;
            float b2 = bias2 ? bias2[cc] : 0.f;
            for (int v = 0; v < 8; v++) {
                int rr = m0 + wM * 32 + sm * 16 + half * 8 + v;
                if (rr >= M) continue;
                float val = (acc[sm][sn][v] + bv) * alpha;
                if (scale2) val = val * s2 + b2;
                if (res)    val += res[(long long)rr * ldr + cc];
                C[(long long)rr * ldc + cc] = act_apply(val, act);
            }
        }
}

// ---------------------------------------------------------------------------
// Utility kernels
// ---------------------------------------------------------------------------
__global__ void k_slice(const float* __restrict__ data, float* __restrict__ out, int ch) {
    long long idx = (long long)blockIdx.x * 256 + threadIdx.x;
    if (idx >= (long long)NNODE * EMB) return;
    int e = (int)(idx % EMB); long long n = idx / EMB;
    int b = (int)(n / LSEG), i = (int)(n % LSEG);
    out[idx] = data[(((long long)(b * 3 + ch) * SEQL) + (i + 1)) * EMB + e];
}

__global__ __launch_bounds__(256) void k_segsum(const float* __restrict__ x, const int* __restrict__ ei,
                                                float* __restrict__ agg, int C) {
    int e = blockIdx.x;
    int c = blockIdx.y * 1024 + threadIdx.x * 4;
    int src = ei[e], dst = ei[NED + e];
    const float4 v = *reinterpret_cast<const float4*>(&x[(long long)src * C + c]);
    float* a = agg + (long long)dst * C + c;
    atomicAdd(a + 0, v.x); atomicAdd(a + 1, v.y); atomicAdd(a + 2, v.z); atomicAdd(a + 3, v.w);
}

__global__ void k_gin_pre(const float* __restrict__ x, const float* __restrict__ agg,
                          const float* __restrict__ eps, float* __restrict__ z, long long total) {
    long long i = (long long)blockIdx.x * 256 + threadIdx.x;
    if (i >= total) return;
    z[i] = (1.f + eps[0]) * x[i] + agg[i];
}

__global__ void k_sub(const float* __restrict__ a, const float* __restrict__ b,
                      float* __restrict__ o, long long total) {
    long long i = (long long)blockIdx.x * 256 + threadIdx.x;
    if (i >= total) return;
    o[i] = a[i] - b[i];
}

__global__ void k_silu_ew(const float* __restrict__ x, float* __restrict__ o, long long total) {
    long long i = (long long)blockIdx.x * 256 + threadIdx.x;
    if (i >= total) return;
    float v = x[i]; o[i] = v / (1.f + expf(-v));
}

__global__ void k_gmp(const float* __restrict__ x, float* __restrict__ out, int C) {
    int idx = blockIdx.x * 256 + threadIdx.x;
    if (idx >= 16 * C) return;
    int b = idx / C, c = idx % C;
    const float* p = x + (long long)b * LSEG * C + c;
    float s = 0.f;
    for (int i = 0; i < LSEG; i++) s += p[(long long)i * C];
    out[idx] = s * (1.0f / LSEG);
}

__global__ void k_stack2(const float* __restrict__ a, const float* __restrict__ b,
                         float* __restrict__ o) {
    int i = blockIdx.x * 256 + threadIdx.x;
    if (i >= 16 * 1024) return;
    o[2 * i] = a[i]; o[2 * i + 1] = b[i];
}

// x[B,S,2]: divide by mean over S of relu(x), per (b, c)
__global__ __launch_bounds__(256) void k_norm_mid2(float* __restrict__ x, int S) {
    __shared__ float r0[256], r1[256];
    int b = blockIdx.x, tid = threadIdx.x;
    float s0 = 0.f, s1 = 0.f;
    for (int s = tid; s < S; s += 256) {
        s0 += fmaxf(x[((long long)b * S + s) * 2], 0.f);
        s1 += fmaxf(x[((long long)b * S + s) * 2 + 1], 0.f);
    }
    r0[tid] = s0; r1[tid] = s1; __syncthreads();
    for (int w = 128; w > 0; w >>= 1) {
        if (tid < w) { r0[tid] += r0[tid + w]; r1[tid] += r1[tid + w]; } __syncthreads();
    }
    float i0 = (float)S / r0[0], i1 = (float)S / r1[0];
    for (int s = tid; s < S; s += 256) {
        x[((long long)b * S + s) * 2]     *= i0;
        x[((long long)b * S + s) * 2 + 1] *= i1;
    }
}

// cls rows [48,1024]: x = x / mean(relu(row)) * 2
__global__ __launch_bounds__(256) void k_cls_norm(float* __restrict__ x) {
    __shared__ float rs[256];
    int r = blockIdx.x, tid = threadIdx.x;
    float* p = x + (long long)r * 1024;
    float s = 0.f;
    for (int c = tid; c < 1024; c += 256) s += fmaxf(p[c], 0.f);
    rs[tid] = s; __syncthreads();
    for (int w = 128; w > 0; w >>= 1) { if (tid < w) rs[tid] += rs[tid + w]; __syncthreads(); }
    float sc = 2.0f * 1024.0f / rs[0];
    for (int c = tid; c < 1024; c += 256) p[c] *= sc;
}

// single-head attention, Em = 2 (gp/pl/cp)
__global__ __launch_bounds__(256) void k_tiny_mha2(
    const float* __restrict__ x, const float* __restrict__ Wi, const float* __restrict__ bi,
    const float* __restrict__ Wo, const float* __restrict__ bo,
    float* __restrict__ out, int S)
{
    __shared__ float sq[2 * 2048], sk[2 * 2048], sv[2 * 2048];
    int b = blockIdx.x, tid = threadIdx.x;
    float wi[12], bia[6], wo[4], boo[2];
    for (int i = 0; i < 12; i++) wi[i] = Wi[i];
    for (int i = 0; i < 6;  i++) bia[i] = bi[i];
    for (int i = 0; i < 4;  i++) wo[i] = Wo[i];
    boo[0] = bo[0]; boo[1] = bo[1];
    for (int s = tid; s < S; s += 256) {
        float x0 = x[((long long)b * S + s) * 2], x1 = x[((long long)b * S + s) * 2 + 1];
        sq[2*s]   = wi[0]*x0 + wi[1]*x1 + bia[0];
        sq[2*s+1] = wi[2]*x0 + wi[3]*x1 + bia[1];
        sk[2*s]   = wi[4]*x0 + wi[5]*x1 + bia[2];
        sk[2*s+1] = wi[6]*x0 + wi[7]*x1 + bia[3];
        sv[2*s]   = wi[8]*x0 + wi[9]*x1 + bia[4];
        sv[2*s+1] = wi[10]*x0 + wi[11]*x1 + bia[5];
    }
    __syncthreads();
    const float sc = 0.70710678118f; // 1/sqrt(hd=2)
    for (int q = tid; q < S; q += 256) {
        float q0 = sq[2*q], q1 = sq[2*q+1];
        float m = -3.4e38f, l = 0.f, o0 = 0.f, o1 = 0.f;
        for (int j = 0; j < S; j++) {
            float s0 = (q0 * sk[2*j] + q1 * sk[2*j+1]) * sc;
            float nm = fmaxf(m, s0);
            float cr = expf(m - nm);
            float p  = expf(s0 - nm);
            l = l * cr + p; o0 = o0 * cr + p * sv[2*j]; o1 = o1 * cr + p * sv[2*j+1];
            m = nm;
        }
        o0 /= l; o1 /= l;
        out[((long long)b * S + q) * 2]     = wo[0]*o0 + wo[1]*o1 + boo[0];
        out[((long long)b * S + q) * 2 + 1] = wo[2]*o0 + wo[3]*o1 + boo[1];
    }
}

__global__ void k_build_ab(const float* __restrict__ cls, float* __restrict__ ab) {
    int idx = blockIdx.x * 256 + threadIdx.x;
    if (idx >= 16 * 2048 * 2) return;
    int c = idx & 1; int t = idx >> 1;
    int r = t % 2048, b = t / 2048;
    int g = r >> 10, e = r & 1023;
    int src = (c == 0) ? 0 : (g == 0 ? 1 : 2);
    ab[idx] = cls[(long long)(b * 3 + src) * 1024 + e];
}

__global__ void k_transpose_pq(const float* __restrict__ ab2, float* __restrict__ pq) {
    int idx = blockIdx.x * 256 + threadIdx.x;
    if (idx >= 16 * 512 * 2) return;
    int c = idx & 1; int t = idx >> 1;
    int s = t % 512, b = t / 512;
    pq[idx] = ab2[(long long)(b * 2 + c) * 512 + s];
}

__global__ __launch_bounds__(256) void k_layernorm(
    const float* __restrict__ a, const float* __restrict__ b,
    const float* __restrict__ g, const float* __restrict__ beta,
    float* __restrict__ out, int C, float eps, int accum)
{
    __shared__ float rs[256], rq[256];
    long long row = blockIdx.x; int tid = threadIdx.x;
    const float* pa = a + row * C;
    const float* pb = b ? b + row * C : nullptr;
    float s = 0.f, q = 0.f;
    for (int c = tid; c < C; c += 256) { float v = pa[c] + (pb ? pb[c] : 0.f); s += v; q += v * v; }
    rs[tid] = s; rq[tid] = q; __syncthreads();
    for (int w = 128; w > 0; w >>= 1) {
        if (tid < w) { rs[tid] += rs[tid + w]; rq[tid] += rq[tid + w]; } __syncthreads();
    }
    float mean = rs[0] / C;
    float var  = rq[0] / C - mean * mean;
    float rstd = rsqrtf(var + eps);
    float* po = out + row * C;
    for (int c = tid; c < C; c += 256) {
        float v = pa[c] + (pb ? pb[c] : 0.f);
        float y = (v - mean) * rstd * g[c] + beta[c];
        po[c] = accum ? po[c] + y : y;
    }
}

__global__ __launch_bounds__(256) void k_softmax_rows(float* __restrict__ x, int W) {
    __shared__ float red[256];
    long long row = blockIdx.x; int tid = threadIdx.x;
    float* p = x + row * W;
    float m = -3.4e38f;
    for (int c = tid; c < W; c += 256) m = fmaxf(m, p[c]);
    red[tid] = m; __syncthreads();
    for (int w = 128; w > 0; w >>= 1) { if (tid < w) red[tid] = fmaxf(red[tid], red[tid + w]); __syncthreads(); }
    m = red[0]; __syncthreads();
    float s = 0.f;
    for (int c = tid; c < W; c += 256) { float e = expf(p[c] - m); p[c] = e; s += e; }
    red[tid] = s; __syncthreads();
    for (int w = 128; w > 0; w >>= 1) { if (tid < w) red[tid] += red[tid + w]; __syncthreads(); }
    float inv = 1.f / red[0];
    for (int c = tid; c < W; c += 256) p[c] *= inv;
}

__global__ __launch_bounds__(128) void k_rowdot(const float* __restrict__ x, const float* __restrict__ w,
                                                float* __restrict__ out, int C) {
    __shared__ float red[128];
    long long row = blockIdx.x; int tid = threadIdx.x;
    const float* p = x + row * C;
    float s = 0.f;
    for (int c = tid; c < C; c += 128) s += p[c] * w[c];
    red[tid] = s; __syncthreads();
    for (int ww = 64; ww > 0; ww >>= 1) { if (tid < ww) red[tid] += red[tid + ww]; __syncthreads(); }
    if (!tid) out[row] = red[0];
}

__global__ __launch_bounds__(256) void k_seqpool(const float* __restrict__ t, const float* __restrict__ logits,
                                                 float* __restrict__ out) {
    __shared__ float wgt[SEQL];
    __shared__ float red[256];
    int b = blockIdx.x, tid = threadIdx.x;
    float m = -3.4e38f;
    for (int s = tid; s < SEQL; s += 256) m = fmaxf(m, logits[b * SEQL + s]);
    red[tid] = m; __syncthreads();
    for (int w = 128; w > 0; w >>= 1) { if (tid < w) red[tid] = fmaxf(red[tid], red[tid + w]); __syncthreads(); }
    m = red[0]; __syncthreads();
    float ss = 0.f;
    for (int s = tid; s < SEQL; s += 256) { float e = expf(logits[b * SEQL + s] - m); wgt[s] = e; ss += e; }
    red[tid] = ss; __syncthreads();
    for (int w = 128; w > 0; w >>= 1) { if (tid < w) red[tid] += red[tid + w]; __syncthreads(); }
    float inv = 1.f / red[0]; __syncthreads();
    for (int s = tid; s < SEQL; s += 256) wgt[s] *= inv;
    __syncthreads();
    for (int c = tid; c < EMB; c += 256) {
        float acc = 0.f;
        for (int s = 0; s < SEQL; s++) acc += wgt[s] * t[((long long)(b * SEQL + s)) * EMB + c];
        out[(long long)b * EMB + c] = acc;
    }
}

__global__ void k_conv1d(const float* __restrict__ x, long long sxn, long long sxc, long long sxl,
                         const float* __restrict__ W, const float* __restrict__ bias,
                         float* __restrict__ out, int Bn, int Cin, int Lin,
                         int Cout, int Ks, int pad, int Lout, int relu) {
    long long idx = (long long)blockIdx.x * 256 + threadIdx.x;
    long long total = (long long)Bn * Cout * Lout;
    if (idx >= total) return;
    int lo = (int)(idx % Lout); long long t = idx / Lout;
    int co = (int)(t % Cout); int n = (int)(t / Cout);
    float acc = bias[co];
    const float* wrow = W + (long long)co * Cin * Ks;
    for (int ci = 0; ci < Cin; ci++) {
        const float* xr = x + n * sxn + ci * sxc;
        const float* wr = wrow + ci * Ks;
        for (int kx = 0; kx < Ks; kx++) {
            int li = lo - pad + kx;
            if (li >= 0 && li < Lin) acc += xr[(long long)li * sxl] * wr[kx];
        }
    }
    if (relu) acc = fmaxf(acc, 0.f);
    out[idx] = acc;
}

__global__ void k_mp2(const float* __restrict__ x, float* __restrict__ out, long long total, int Lh) {
    long long idx = (long long)blockIdx.x * 256 + threadIdx.x;
    if (idx >= total) return;
    long long rc = idx / Lh; int i = (int)(idx % Lh);
    out[idx] = fmaxf(x[rc * 2ll * Lh + 2 * i], x[rc * 2ll * Lh + 2 * i + 1]);
}

// in: flat [48,1024]  ->  out: [32,128,16]
__global__ void k_pair(const float* __restrict__ in, float* __restrict__ out) {
    int idx = blockIdx.x * 256 + threadIdx.x;
    if (idx >= 32 * 128 * 16) return;
    int c16 = idx % 16; int t = idx / 16;
    int r = t % 128; int bj = t / 128;
    int b = bj / 2, j = bj % 2;
    int src = (r < 64) ? 0 : (j == 0 ? 1 : 2);
    int rr = r & 63;
    out[idx] = in[(long long)(b * 3 + src) * 1024 + c16 * 64 + rr];
}

__global__ void k_concat(const float* __restrict__ rout, const float* __restrict__ ctx,
                         const float* __restrict__ glob, const float* __restrict__ gnn,
                         float* __restrict__ o) {
    int idx = blockIdx.x * 256 + threadIdx.x;
    if (idx >= 16 * 6144) return;
    int b = idx / 6144, c = idx % 6144;
    float v;
    if      (c < 2048) v = rout[(long long)b * 2048 + c];
    else if (c < 4096) v = ctx [(long long)b * 2048 + (c - 2048)];
    else if (c < 5120) v = glob[(long long)b * 1024 + (c - 4096)];
    else               v = gnn [(long long)b * 1024 + (c - 5120)];
    o[idx] = v;
}

// ---------------------------------------------------------------------------
// Host side
// ---------------------------------------------------------------------------
extern "C" void kernel_launch(void* const* d_in, const int* in_sizes, int n_in,
                              void* d_out, int out_size, void* d_ws, size_t ws_size,
                              hipStream_t stream) {
    (void)in_sizes; (void)n_in; (void)out_size; (void)ws_size;
    #define PF(i) ((const float*)d_in[i])
    const float* data = PF(0);
    const int*   eidx = (const int*)d_in[1];
    // param indices in setup_inputs() dict (insertion) order, leaves flattened
    // 4..6 gcn1{W,b,eps} 7..9 gcn2 10..12 gcn3 13..15 gcnr 16..17 fc_post
    // 18..21 gp_att 22..25 pl_att 26..29 cp_att 30..31 fc_g1 32..33 fc_g2
    // 34..35 fc_res 36..39 n1g n1b n2g n2b 40..43 trans.att 44..47 g1 b1 g2 b2
    // 48..51 ff1 ff2 52 tq 53..56 ctx1 ctx2 57..64 res1..4 65..68 pr1 pr2
    // 69..70 pooler 71..72 fc_ab 73..74 bn_g bn_b 75..80 f1 f2 f3

    // ---- bump allocator over d_ws (deterministic layout; ~0.95 GB peak)
    size_t off = 0;
    auto alloc = [&](size_t nfl) -> float* {
        float* p = (float*)((char*)d_ws + off);
        off += ((nfl * sizeof(float) + 255) / 256) * 256;
        return p;
    };
    auto nb = [](long long n) { return (unsigned)((n + 255) / 256); };

    // smalls (persistent)
    float* s_gw4  = alloc(16 * 1024);
    float* s_gm4  = alloc(16 * 1024);
    float* s_gwm  = alloc(16 * 1024 * 2);
    float* s_gwmo = alloc(16 * 1024 * 2);
    float* s_gnn0 = alloc(16 * 1024);
    float* s_resv = alloc(16 * 1024);
    float* s_gln  = alloc(16 * 1024);
    float* s_gsil = alloc(16 * 1024);
    float* s_gnn  = alloc(16 * 1024);
    float* s_cls  = alloc(48 * 1024);
    float* s_ab   = alloc(16 * 2048 * 2);
    float* s_abo  = alloc(16 * 2048 * 2);
    float* s_ab2  = alloc(32 * 512);
    float* s_pq   = alloc(16 * 512 * 2);
    float* s_glob = alloc(16 * 1024);
    float* s_logi = alloc(NTOK);
    float* s_tpl  = alloc(48 * 1024);
    float* s_p1   = alloc(32 * 128 * 16);
    float* s_c1   = alloc(32 * 96 * 16);
    float* s_c2   = alloc(32 * 64 * 16);
    float* s_e1   = alloc((size_t)48 * 128 * 500);
    float* s_e2   = alloc((size_t)48 * 80 * 512);
    float* s_e2p  = alloc((size_t)48 * 80 * 256);
    float* s_e3   = alloc((size_t)48 * 40 * 256);
    float* s_e3p  = alloc((size_t)48 * 40 * 128);
    float* s_e4   = alloc((size_t)48 * 16 * 128);
    float* s_e4p  = alloc((size_t)48 * 16 * 64);
    float* s_p2   = alloc(32 * 128 * 16);
    float* s_r1   = alloc(32 * 96 * 16);
    float* s_r2   = alloc(32 * 64 * 16);
    float* s_fcin = alloc(16 * 6144);
    float* s_f1   = alloc(16 * 64);
    float* s_f2   = alloc(16 * 8);
    size_t mark = off;

    // GNN-phase big buffers (recycled for the transformer afterwards)
    float* gw  = alloc((size_t)NNODE * C1);
    float* gm  = alloc((size_t)NNODE * C1);
    float* gw3 = alloc((size_t)NNODE * C1);
    float* gm3 = alloc((size_t)NNODE * C1);
    float* agg = alloc((size_t)NNODE * C2);
    float* zb  = alloc((size_t)NNODE * C2);
    float* hb  = alloc((size_t)NNODE * C1);
    float* h2b = alloc((size_t)NNODE * C2);
    float* h3b = alloc((size_t)NNODE * C1);
    float* rb  = alloc((size_t)NNODE * C1);

    auto gemmS = [&](const float* A, long long lda, const float* W, long long ldw,
                     float* C_, long long ldc, int M, int N, int K,
                     const float* bias, int act,
                     const float* res = nullptr, long long ldr = 0,
                     const float* sc2 = nullptr, const float* b2 = nullptr,
                     float alpha = 1.0f) {
        dim3 g((N + TBN - 1) / TBN, (M + TBM - 1) / TBM, 1);
        k_gemm<1><<<g, 256, 0, stream>>>(A, lda, 0, 0, W, ldw, 0, 0, C_, ldc, 0, 0,
                                         bias, sc2, b2, res, ldr, M, N, K, 1, alpha, act);
    };

    // =========================== GNN branches ===========================
    k_slice<<<nb((long long)NNODE * EMB), 256, 0, stream>>>(data, gw, 1);
    k_slice<<<nb((long long)NNODE * EMB), 256, 0, stream>>>(data, gm, 2);

    auto branch = [&](const float* x, float* out3) {
        long long nc1 = (long long)NNODE * C1, nc2 = (long long)NNODE * C2;
        (void)hipMemsetAsync(agg, 0, nc1 * sizeof(float), stream);
        k_segsum<<<dim3(NED, 1), 256, 0, stream>>>(x, eidx, agg, C1);
        k_gin_pre<<<nb(nc1), 256, 0, stream>>>(x, agg, PF(6), zb, nc1);
        gemmS(zb, C1, PF(4), C1, hb, C1, NNODE, C1, C1, PF(5), A_SILU2);        // h
        k_gin_pre<<<nb(nc1), 256, 0, stream>>>(x, agg, PF(15), zb, nc1);        // reuse agg(x)
        gemmS(zb, C1, PF(13), C1, rb, C1, NNODE, C1, C1, PF(14), A_SILU2);      // silu(gin_r)
        (void)hipMemsetAsync(agg, 0, nc1 * sizeof(float), stream);
        k_segsum<<<dim3(NED, 1), 256, 0, stream>>>(hb, eidx, agg, C1);
        k_gin_pre<<<nb(nc1), 256, 0, stream>>>(hb, agg, PF(9), zb, nc1);
        gemmS(zb, C1, PF(7), C1, h2b, C2, NNODE, C2, C1, PF(8), A_SILU2);       // h2
        (void)hipMemsetAsync(agg, 0, nc2 * sizeof(float), stream);
        k_segsum<<<dim3(NED, 2), 256, 0, stream>>>(h2b, eidx, agg, C2);
        k_gin_pre<<<nb(nc2), 256, 0, stream>>>(h2b, agg, PF(12), zb, nc2);
        gemmS(zb, C2, PF(10), C2, h3b, C1, NNODE, C1, C2, PF(11), A_SILU);      // h3
        gemmS(rb, C1, PF(16), C1, out3, C1, NNODE, C1, C1, PF(17), A_SILU, h3b, C1);
    };
    branch(gw, gw3);
    branch(gm, gm3);

    k_gmp<<<nb(16 * C1), 256, 0, stream>>>(gw3, s_gw4, C1);
    k_gmp<<<nb(16 * C1), 256, 0, stream>>>(gm3, s_gm4, C1);
    k_stack2<<<nb(16 * 1024), 256, 0, stream>>>(s_gw4, s_gm4, s_gwm);
    k_norm_mid2<<<16, 256, 0, stream>>>(s_gwm, 1024);
    k_tiny_mha2<<<16, 256, 0, stream>>>(s_gwm, PF(18), PF(19), PF(20), PF(21), s_gwmo, 1024);
    gemmS(s_gwmo, 2048, PF(30), 2048, s_gnn0, 1024, 16, 1024, 2048, PF(31), A_NONE);

    k_sub<<<nb((long long)NNODE * C1), 256, 0, stream>>>(gw3, gm3, hb, (long long)NNODE * C1);
    gemmS(hb, C1, PF(34), C1, zb, C1, NNODE, C1, C1, PF(35), A_NONE);
    k_gmp<<<nb(16 * C1), 256, 0, stream>>>(zb, s_resv, C1);

    k_layernorm<<<16, 256, 0, stream>>>(s_gnn0, nullptr, PF(36), PF(37), s_gln, 1024, 1e-6f, 0);
    k_layernorm<<<16, 256, 0, stream>>>(s_resv, nullptr, PF(38), PF(39), s_gln, 1024, 1e-6f, 1);
    k_silu_ew<<<nb(16 * 1024), 256, 0, stream>>>(s_gln, s_gsil, 16 * 1024);
    gemmS(s_gsil, 1024, PF(32), 1024, s_gnn, 1024, 16, 1024, 1024, PF(33), A_NONE);

    // =========================== CLS / pooler path ===========================
    gemmS(data, (long long)SEQL * EMB, PF(69), 1024, s_cls, 1024, 48, 1024, 1024, PF(70), A_TANH);
    k_cls_norm<<<48, 256, 0, stream>>>(s_cls);
    k_build_ab<<<nb(16 * 2048 * 2), 256, 0, stream>>>(s_cls, s_ab);
    k_tiny_mha2<<<16, 256, 0, stream>>>(s_ab, PF(22), PF(23), PF(24), PF(25), s_abo, 2048);
    float abAlpha = 1.0f / sqrtf(1.0f + 1e-5f);
    gemmS(s_abo, 2048, PF(71), 2048, s_ab2, 512, 32, 512, 2048, PF(72), A_RELU,
          nullptr, 0, PF(73), PF(74), abAlpha);
    k_transpose_pq<<<nb(16 * 512 * 2), 256, 0, stream>>>(s_ab2, s_pq);
    k_norm_mid2<<<16, 256, 0, stream>>>(s_pq, 512);
    k_tiny_mha2<<<16, 256, 0, stream>>>(s_pq, PF(26), PF(27), PF(28), PF(29), s_glob, 512);

    // =========================== Transformer (recycle big arena) ===========================
    off = mark;
    float* qkv    = alloc((size_t)NTOK * 3072);
    float* scores = alloc((size_t)192 * 500 * 500);
    float* ob     = alloc((size_t)NTOK * 1024);
    float* tb     = alloc((size_t)NTOK * 1024);
    float* ffb    = alloc((size_t)NTOK * 2048);
    float* attnout = scores;  // scores dead after att@v
    float* f2out   = ob;      // o dead after projection
    float* t2     = scores;   // attnout dead after LN1

    gemmS(data, EMB, PF(40), EMB, qkv, 3072, NTOK, 3072, EMB, PF(41), A_NONE);
    {   // scores = q @ k^T / 16   (192 batches = 48 x 4 heads)
        dim3 g((500 + TBN - 1) / TBN, (500 + TBM - 1) / TBM, 192);
        k_gemm<1><<<g, 256, 0, stream>>>(qkv, 3072, 500ll * 3072, 256,
                                         qkv + 1024, 3072, 500ll * 3072, 256,
                                         scores, 500, 4ll * 250000, 250000,
                                         nullptr, nullptr, nullptr, nullptr, 0,
                                         500, 500, 256, 4, 0.0625f, A_NONE);
    }
    k_softmax_rows<<<192 * 500, 256, 0, stream>>>(scores, 500);
    {   // o = att @ v
        dim3 g((256 + TBN - 1) / TBN, (500 + TBM - 1) / TBM, 192);
        k_gemm<0><<<g, 256, 0, stream>>>(scores, 500, 4ll * 250000, 250000,
                                         qkv + 2048, 3072, 500ll * 3072, 256,
                                         ob, 1024, 500ll * 1024, 256,
                                         nullptr, nullptr, nullptr, nullptr, 0,
                                         500, 256, 500, 4, 1.0f, A_NONE);
    }
    gemmS(ob, 1024, PF(42), 1024, attnout, 1024, NTOK, 1024, 1024, PF(43), A_NONE);
    k_layernorm<<<NTOK, 256, 0, stream>>>(data, attnout, PF(44), PF(45), tb, EMB, 1e-5f, 0);
    gemmS(tb, 1024, PF(48), 1024, ffb, 2048, NTOK, 2048, 1024, PF(49), A_RELU);
    gemmS(ffb, 2048, PF(50), 2048, f2out, 1024, NTOK, 1024, 2048, PF(51), A_NONE);
    k_layernorm<<<NTOK, 256, 0, stream>>>(tb, f2out, PF(46), PF(47), t2, EMB, 1e-5f, 0);
    k_rowdot<<<NTOK, 128, 0, stream>>>(t2, PF(52), s_logi, EMB);
    k_seqpool<<<NB3, 256, 0, stream>>>(t2, s_logi, s_tpl);

    // =========================== Conv paths ===========================
    k_pair<<<nb(32 * 128 * 16), 256, 0, stream>>>(s_tpl, s_p1);
    k_conv1d<<<nb(32ll * 96 * 16), 256, 0, stream>>>(s_p1, 128 * 16, 16, 1, PF(53), PF(54),
                                                     s_c1, 32, 128, 16, 96, 1, 0, 16, 1);
    k_conv1d<<<nb(32ll * 64 * 16), 256, 0, stream>>>(s_c1, 96 * 16, 16, 1, PF(55), PF(56),
                                                     s_c2, 32, 96, 16, 64, 1, 0, 16, 1);

    k_conv1d<<<nb(48ll * 128 * 500), 256, 0, stream>>>(data, (long long)SEQL * EMB, 1, EMB,
                                                       PF(57), PF(58), s_e1, 48, 1024, 500, 128, 3, 1, 500, 1);
    k_conv1d<<<nb(48ll * 80 * 512), 256, 0, stream>>>(s_e1, 128ll * 500, 500, 1,
                                                      PF(59), PF(60), s_e2, 48, 128, 500, 80, 49, 30, 512, 1);
    k_mp2<<<nb(48ll * 80 * 256), 256, 0, stream>>>(s_e2, s_e2p, 48ll * 80 * 256, 256);
    k_conv1d<<<nb(48ll * 40 * 256), 256, 0, stream>>>(s_e2p, 80ll * 256, 256, 1,
                                                      PF(61), PF(62), s_e3, 48, 80, 256, 40, 31, 15, 256, 1);
    k_mp2<<<nb(48ll * 40 * 128), 256, 0, stream>>>(s_e3, s_e3p, 48ll * 40 * 128, 128);
    k_conv1d<<<nb(48ll * 16 * 128), 256, 0, stream>>>(s_e3p, 40ll * 128, 128, 1,
                                                      PF(63), PF(64), s_e4, 48, 40, 128, 16, 15, 7, 128, 1);
    k_mp2<<<nb(48ll * 16 * 64), 256, 0, stream>>>(s_e4, s_e4p, 48ll * 16 * 64, 64);

    k_pair<<<nb(32 * 128 * 16), 256, 0, stream>>>(s_e4p, s_p2);
    k_conv1d<<<nb(32ll * 96 * 16), 256, 0, stream>>>(s_p2, 128 * 16, 16, 1, PF(65), PF(66),
                                                     s_r1, 32, 128, 16, 96, 1, 0, 16, 1);
    k_conv1d<<<nb(32ll * 64 * 16), 256, 0, stream>>>(s_r1, 96 * 16, 16, 1, PF(67), PF(68),
                                                     s_r2, 32, 96, 16, 64, 1, 0, 16, 1);

    // =========================== Final MLP ===========================
    k_concat<<<nb(16 * 6144), 256, 0, stream>>>(s_r2, s_c2, s_glob, s_gnn, s_fcin);
    gemmS(s_fcin, 6144, PF(75), 6144, s_f1, 64, 16, 64, 6144, PF(76), A_RELU);
    gemmS(s_f1, 64, PF(77), 64, s_f2, 8, 16, 8, 64, PF(78), A_RELU);
    gemmS(s_f2, 8, PF(79), 8, (float*)d_out, 1, 16, 1, 8, PF(80), A_NONE);
    #undef PF
}